// Model_51453708206382
// MI455X (gfx1250) — compile-verified
//
#include <hip/hip_runtime.h>

// Problem constants (match reference)
constexpr int B = 4, NQ = 32, NKV = 8, S = 2048, D = 128, MAXPOS = 8192;

#define AS3 __attribute__((address_space(3)))

// native clang vector type: required by __builtin_nontemporal_load/store
typedef float v4f __attribute__((ext_vector_type(4)));

// ---------------------------------------------------------------------------
// Bulk cache passthrough via CDNA5 async global<->LDS DMA path (ASYNCcnt).
// Each wave owns a private 4KB LDS slab; 8 x B128 async loads stage 4KB,
// s_wait_asynccnt 0, then 8 x B128 async stores drain it. The immediate
// offset is applied to both the LDS and global addresses (ISA 08, §4.4),
// so one base VGPR per direction suffices.
// ---------------------------------------------------------------------------
__global__ __launch_bounds__(256) void cache_copy_async(const float* __restrict__ src,
                                                        float* __restrict__ dst,
                                                        size_t nbytes) {
  __shared__ char smem[8][4096];
  const int lane = threadIdx.x & 31;
  const int wave = threadIdx.x >> 5;
  const unsigned lds_off =
      (unsigned)(unsigned long long)(AS3 char*)&smem[wave][0] + (unsigned)lane * 16u;

  const size_t TILE = 32 * 1024;  // bytes per block per iteration
  for (size_t base = (size_t)blockIdx.x * TILE; base < nbytes;
       base += (size_t)gridDim.x * TILE) {
    const size_t woff = base + (size_t)wave * 4096 + (size_t)lane * 16;
    unsigned long long gsrc = (unsigned long long)src + woff;
    unsigned long long gdst = (unsigned long long)dst + woff;

    // previous iteration's async stores must have drained this LDS slab
    asm volatile("s_wait_asynccnt 0x0" ::: "memory");
#pragma unroll
    for (int c = 0; c < 8; ++c)
      asm volatile("global_load_async_to_lds_b128 %0, %1, off offset:%c2"
                   :: "v"(lds_off), "v"(gsrc), "i"(c * 512) : "memory");
    asm volatile("s_wait_asynccnt 0x0" ::: "memory");
#pragma unroll
    for (int c = 0; c < 8; ++c)
      asm volatile("global_store_async_from_lds_b128 %0, %1, off offset:%c2"
                   :: "v"(gdst), "v"(lds_off), "i"(c * 512) : "memory");
  }
  asm volatile("s_wait_asynccnt 0x0" ::: "memory");
}

// ---------------------------------------------------------------------------
// Precompute cos/sin table: [B][S][64] each. Reused by all 40 heads, so the
// expensive (accurate) sincos runs 524K times instead of 42M times.
// ---------------------------------------------------------------------------
__global__ __launch_bounds__(256) void rope_table(const int* __restrict__ pos_ids,    // [B,S]
                                                  const float* __restrict__ inv_freq, // [64]
                                                  float* __restrict__ cos_tab,        // [B*S*64]
                                                  float* __restrict__ sin_tab) {
  const int i = blockIdx.x * blockDim.x + threadIdx.x;
  const int total = B * S * (D / 2);
  if (i >= total) return;
  const int fi = i & 63;
  const int bs = i >> 6;  // b*S + s
  const float f = (float)pos_ids[bs] * inv_freq[fi];
  float sn, cs;
  sincosf(f, &sn, &cs);
  cos_tab[i] = cs;
  sin_tab[i] = sn;
}

// ---------------------------------------------------------------------------
// RMS-norm + RoPE, one wave32 per 128-float row. Lane l holds elements
// [4l, 4l+3]; rotate_half partner lane is l^16 (64 elements = 16 lanes);
// sign is -1 for lanes 0..15 (the "-x[d:]" half). Optionally scatters the
// rotated row into the cache at row cache_pos[s]. Big streams use NT policy
// so the 4MB rope table + weights stay hot in L2.
// ---------------------------------------------------------------------------
__global__ __launch_bounds__(256) void rmsnorm_rope(const float* __restrict__ x,        // [B,H,S,D]
                                                    const float* __restrict__ w,        // [D]
                                                    const int* __restrict__ pos_ids,    // [B,S]
                                                    const float* __restrict__ inv_freq, // [64]
                                                    const float* __restrict__ cos_tab,  // [B*S*64] or null
                                                    const float* __restrict__ sin_tab,  // [B*S*64] or null
                                                    const int* __restrict__ cache_pos,  // [S] or null
                                                    float* __restrict__ out,            // [B,H,S,D]
                                                    float* __restrict__ cache_out,      // [B,H,MAXPOS,D] or null
                                                    int H,
                                                    const float* __restrict__ eps_p) {
  const int lane = threadIdx.x & 31;
  const int wave = threadIdx.x >> 5;
  const size_t row = (size_t)blockIdx.x * 8 + wave;
  const size_t nrows = (size_t)B * H * S;
  if (row >= nrows) return;

  const int s = (int)(row % S);
  const size_t bh = row / S;            // b*H + h
  const int b = (int)(bh / H);
  const float eps = *eps_p;

  const v4f v = __builtin_nontemporal_load((const v4f*)x + row * (D / 4) + lane);
  float ss = v.x * v.x + v.y * v.y + v.z * v.z + v.w * v.w;
#pragma unroll
  for (int m = 16; m; m >>= 1) ss += __shfl_xor(ss, m, 32);
  const float scale = rsqrtf(ss * (1.0f / (float)D) + eps);

  const v4f wt = ((const v4f*)w)[lane];
  v4f n;
  n.x = v.x * scale * wt.x;
  n.y = v.y * scale * wt.y;
  n.z = v.z * scale * wt.z;
  n.w = v.w * scale * wt.w;

  // rotate_half: partner lane holds the +/-64 element block
  v4f p;
  p.x = __shfl_xor(n.x, 16, 32);
  p.y = __shfl_xor(n.y, 16, 32);
  p.z = __shfl_xor(n.z, 16, 32);
  p.w = __shfl_xor(n.w, 16, 32);
  const float sgn = (lane < 16) ? -1.0f : 1.0f;

  v4f c4, s4;
  if (cos_tab != nullptr) {
    const size_t tbase = ((size_t)b * S + s) * 16;   // 16 v4f per (b,s)
    c4 = ((const v4f*)cos_tab)[tbase + (lane & 15)];
    s4 = ((const v4f*)sin_tab)[tbase + (lane & 15)];
  } else {
    const float pos = (float)pos_ids[(size_t)b * S + s];
    const int fbase = (lane & 15) * 4;
    float sn, cs;
    __sincosf(pos * inv_freq[fbase + 0], &sn, &cs); s4.x = sn; c4.x = cs;
    __sincosf(pos * inv_freq[fbase + 1], &sn, &cs); s4.y = sn; c4.y = cs;
    __sincosf(pos * inv_freq[fbase + 2], &sn, &cs); s4.z = sn; c4.z = cs;
    __sincosf(pos * inv_freq[fbase + 3], &sn, &cs); s4.w = sn; c4.w = cs;
  }

  v4f r;
  r.x = n.x * c4.x + sgn * p.x * s4.x;
  r.y = n.y * c4.y + sgn * p.y * s4.y;
  r.z = n.z * c4.z + sgn * p.z * s4.z;
  r.w = n.w * c4.w + sgn * p.w * s4.w;

  __builtin_nontemporal_store(r, (v4f*)out + row * (D / 4) + lane);

  if (cache_out != nullptr) {
    const int cp = cache_pos[s];
    __builtin_nontemporal_store(r, (v4f*)cache_out + (bh * MAXPOS + cp) * (D / 4) + lane);
  }
}

// ---------------------------------------------------------------------------
// value -> value_cache_out row scatter (v4f granularity)
// ---------------------------------------------------------------------------
__global__ __launch_bounds__(256) void value_scatter(const float* __restrict__ val,     // [B,NKV,S,D]
                                                     const int* __restrict__ cache_pos, // [S]
                                                     float* __restrict__ vco) {         // [B,NKV,MAXPOS,D]
  const size_t i = (size_t)blockIdx.x * blockDim.x + threadIdx.x;  // v4f index
  const size_t total = (size_t)B * NKV * S * (D / 4);
  if (i >= total) return;
  const size_t row = i / (D / 4);
  const int c4 = (int)(i % (D / 4));
  const int s = (int)(row % S);
  const size_t bh = row / S;
  const int cp = cache_pos[s];
  const v4f v = __builtin_nontemporal_load((const v4f*)val + i);
  __builtin_nontemporal_store(v, (v4f*)vco + (bh * MAXPOS + cp) * (D / 4) + c4);
}

extern "C" void kernel_launch(void* const* d_in, const int* in_sizes, int n_in,
                              void* d_out, int out_size, void* d_ws, size_t ws_size,
                              hipStream_t stream) {
  const float* query        = (const float*)d_in[0];
  const float* key          = (const float*)d_in[1];
  const float* value        = (const float*)d_in[2];
  const int*   position_ids = (const int*)d_in[3];
  const float* key_cache    = (const float*)d_in[4];
  const float* value_cache  = (const float*)d_in[5];
  const int*   cache_pos    = (const int*)d_in[6];
  const float* q_w          = (const float*)d_in[7];
  const float* k_w          = (const float*)d_in[8];
  const float* inv_freq     = (const float*)d_in[9];
  const float* eps_p        = (const float*)d_in[10];

  float* q_rot = (float*)d_out;                          // [B,NQ,S,D]
  float* k_rot = q_rot + (size_t)B * NQ * S * D;         // [B,NKV,S,D]
  float* kco   = k_rot + (size_t)B * NKV * S * D;        // [B,NKV,MAXPOS,D]
  float* vco   = kco   + (size_t)B * NKV * MAXPOS * D;   // [B,NKV,MAXPOS,D]

  const size_t cache_bytes = (size_t)B * NKV * MAXPOS * D * sizeof(float);  // 134 MB

  // cos/sin table in workspace (4 MB); fall back to in-kernel sincos if tight
  const size_t tab_elems = (size_t)B * S * (D / 2);
  float* cos_tab = nullptr;
  float* sin_tab = nullptr;
  if (ws_size >= 2 * tab_elems * sizeof(float)) {
    cos_tab = (float*)d_ws;
    sin_tab = cos_tab + tab_elems;
    rope_table<<<(unsigned)((tab_elems + 255) / 256), 256, 0, stream>>>(
        position_ids, inv_freq, cos_tab, sin_tab);
  }

  // full cache passthrough via async-LDS DMA path (stream-ordered before scatters)
  cache_copy_async<<<4096, 256, 0, stream>>>(key_cache, kco, cache_bytes);
  cache_copy_async<<<4096, 256, 0, stream>>>(value_cache, vco, cache_bytes);

  // q path (no cache scatter). 8 rows/block, B*NQ*S rows.
  rmsnorm_rope<<<(B * NQ * S) / 8, 256, 0, stream>>>(
      query, q_w, position_ids, inv_freq, cos_tab, sin_tab,
      nullptr, q_rot, nullptr, NQ, eps_p);

  // k path + scatter into key cache
  rmsnorm_rope<<<(B * NKV * S) / 8, 256, 0, stream>>>(
      key, k_w, position_ids, inv_freq, cos_tab, sin_tab,
      cache_pos, k_rot, kco, NKV, eps_p);

  // value scatter into value cache
  const size_t vtotal = (size_t)B * NKV * S * (D / 4);
  value_scatter<<<(unsigned)((vtotal + 255) / 256), 256, 0, stream>>>(
      value, cache_pos, vco);
}